// LightGCN_48919677501916
// MI455X (gfx1250) — compile-verified
//
#include <hip/hip_runtime.h>

// LightGCN forward for MI455X (gfx1250, wave32).
// Memory-bound: ~2.7 GB traffic -> ~120us at 23.3 TB/s; all buffers L2-resident (192MB).
// WMMA (V_WMMA_F32_16X16X4_F32) used for the label-pair dot products (diagonal of
// a 16x16 batched A*B), keeping full f32 precision.

typedef __attribute__((ext_vector_type(2))) float v2f;
typedef __attribute__((ext_vector_type(8))) float v8f;

#define ED 64  // EMBED_DIM

__device__ __forceinline__ void atomAddF(float* p, float v) {
  // Relaxed, device-scope float atomic -> native global_atomic_add_f32 (no CAS loop).
  __hip_atomic_fetch_add(p, v, __ATOMIC_RELAXED, __HIP_MEMORY_SCOPE_AGENT);
}

// out = alpha[0]*emb ; bufA = 0 ; deg = 0  (one fused elementwise pass over N*64)
__global__ void lgcn_init(const float* __restrict__ emb, const float* __restrict__ alpha,
                          float* __restrict__ out, float* __restrict__ bufA,
                          float* __restrict__ deg, int N, long n64) {
  float a0 = alpha[0];
  long stride = (long)gridDim.x * blockDim.x;
  for (long i = blockIdx.x * (long)blockDim.x + threadIdx.x; i < n64; i += stride) {
    out[i]  = a0 * emb[i];
    bufA[i] = 0.0f;
    if (i < N) deg[i] = 0.0f;
  }
}

// deg[col[e]] += 1
__global__ void lgcn_deg(const int* __restrict__ col, float* __restrict__ deg, int E) {
  int e = blockIdx.x * blockDim.x + threadIdx.x;
  if (e < E) atomAddF(deg + col[e], 1.0f);
}

// deg -> d^{-1/2} in place
__global__ void lgcn_dinv(float* __restrict__ deg, int N) {
  int i = blockIdx.x * blockDim.x + threadIdx.x;
  if (i < N) {
    float d = deg[i];
    deg[i] = (d > 0.0f) ? rsqrtf(d) : 0.0f;
  }
}

// One propagation layer: xout[col] += norm(row,col) * xin[row]
// 16 lanes per edge, float4 (b128) gather + 4 f32 atomics per lane.
__global__ void lgcn_prop(const int* __restrict__ row, const int* __restrict__ col,
                          const float* __restrict__ dinv, const float* __restrict__ xin,
                          float* __restrict__ xout, int E) {
  long t = blockIdx.x * (long)blockDim.x + threadIdx.x;
  int e = (int)(t >> 4);
  if (e >= E) return;
  int c4 = ((int)t & 15) << 2;
  int r = row[e], c = col[e];
  float nrm = dinv[r] * dinv[c];
  const float4 v = *(const float4*)(xin + (long)r * ED + c4);
  float* dst = xout + (long)c * ED + c4;
  atomAddF(dst + 0, v.x * nrm);
  atomAddF(dst + 1, v.y * nrm);
  atomAddF(dst + 2, v.z * nrm);
  atomAddF(dst + 3, v.w * nrm);
}

// out += alpha[ai] * x ; optionally zero the other ping-pong buffer for next layer
__global__ void lgcn_accum(const float* __restrict__ x, const float* __restrict__ alpha,
                           int ai, float* __restrict__ out, float* __restrict__ zbuf,
                           long n64) {
  float a = alpha[ai];
  long stride = (long)gridDim.x * blockDim.x;
  for (long i = blockIdx.x * (long)blockDim.x + threadIdx.x; i < n64; i += stride) {
    out[i] += a * x[i];
    if (zbuf) zbuf[i] = 0.0f;
  }
}

// Label-edge dot products via WMMA: one wave handles 16 pairs.
// A = 16 src rows (16x64), B = 16 dst vectors as columns (64x16),
// D = A*B accumulated over K in steps of 4 with V_WMMA_F32_16X16X4_F32;
// result = diagonal of the 16x16 f32 accumulator.
// A 16x4 f32 layout: lanes 0-15 hold {K=k,k+1} of row M=lane; lanes 16-31 hold
// {K=k+2,k+3} of row M=lane-16. B mirrors this with N=lane. Symmetric per-lane feed.
__global__ void lgcn_dot_wmma(const int* __restrict__ lab, const float* __restrict__ x,
                              float* __restrict__ out, int L) {
  int wave = (int)((blockIdx.x * (long)blockDim.x + threadIdx.x) >> 5);
  int lane = (int)(threadIdx.x & 31);
  int base = wave * 16;
  if (base >= L) return;  // wave-uniform: EXEC stays all-ones for WMMA
  int p = base + (lane & 15);
  if (p >= L) p = L - 1;  // tail clamp (harmless duplicate loads)
  const float* sv = x + (long)lab[p] * ED;      // src embedding row
  const float* dv = x + (long)lab[L + p] * ED;  // dst embedding row
  int ks = (lane >> 4) << 1;  // 0 for lanes 0-15, 2 for lanes 16-31

  v8f c = {0.f, 0.f, 0.f, 0.f, 0.f, 0.f, 0.f, 0.f};
#pragma unroll
  for (int k = 0; k < ED; k += 4) {
    v2f a = *(const v2f*)(sv + k + ks);
    v2f b = *(const v2f*)(dv + k + ks);
    // 8 args: (neg_a, A, neg_b, B, c_mod, C, reuse_a, reuse_b)
    c = __builtin_amdgcn_wmma_f32_16x16x4_f32(false, a, false, b, (short)0, c,
                                              false, false);
  }

  // Diagonal extraction per C/D layout:
  //   i in [0,8):  lane i,      VGPR i       -> out[base+i]
  //   i in [8,16): lane 16+i, VGPR i-8     -> lanes 24..31, VGPR lane-24
  if (lane < 8) {
    if (base + lane < L) out[base + lane] = c[lane];
  } else if (lane >= 24) {
    int i = lane - 16;  // 8..15
    if (base + i < L) out[base + i] = c[lane - 24];
  }
}

extern "C" void kernel_launch(void* const* d_in, const int* in_sizes, int n_in,
                              void* d_out, int out_size, void* d_ws, size_t ws_size,
                              hipStream_t stream) {
  const int*   ei    = (const int*)d_in[0];    // edge_index [2,E]
  const int*   eli   = (const int*)d_in[1];    // edge_label_index [2,L]
  const float* emb   = (const float*)d_in[2];  // [N,64] f32
  const float* alpha = (const float*)d_in[3];  // [4] f32

  const int  E   = in_sizes[0] / 2;
  const int  L   = in_sizes[1] / 2;
  const int  N   = in_sizes[2] / ED;
  const long n64 = (long)N * ED;

  // Carve workspace: deg/dinv [N], bufA [N*64], bufB [N*64], out_emb [N*64]
  char* ws = (char*)d_ws;
  size_t off = 0;
  auto carve = [&](size_t bytes) -> void* {
    void* p = ws + off;
    off = (off + bytes + 255) & ~(size_t)255;
    return p;
  };
  float* deg  = (float*)carve((size_t)N * sizeof(float));
  float* bufA = (float*)carve((size_t)n64 * sizeof(float));
  float* bufB = (float*)carve((size_t)n64 * sizeof(float));
  float* oute = (float*)carve((size_t)n64 * sizeof(float));
  (void)ws_size; (void)n_in; (void)out_size;

  dim3 blk(256);
  int gEl = 4096;                                   // grid-stride elementwise
  int gE  = (E + 255) / 256;                        // per-edge
  int gN  = (N + 255) / 256;                        // per-node
  int gEw = (int)(((long)E * 16 + 255) / 256);      // 16 lanes per edge
  int waves = (L + 15) / 16;
  int gD  = (int)(((long)waves * 32 + 255) / 256);  // one wave per 16 pairs

  lgcn_init<<<gEl, blk, 0, stream>>>(emb, alpha, oute, bufA, deg, N, n64);
  lgcn_deg<<<gE, blk, 0, stream>>>(ei + E, deg, E);
  lgcn_dinv<<<gN, blk, 0, stream>>>(deg, N);

  // Layer 1: emb -> bufA ; out += a1*bufA ; zero bufB
  lgcn_prop<<<gEw, blk, 0, stream>>>(ei, ei + E, deg, emb, bufA, E);
  lgcn_accum<<<gEl, blk, 0, stream>>>(bufA, alpha, 1, oute, bufB, n64);
  // Layer 2: bufA -> bufB ; out += a2*bufB ; zero bufA
  lgcn_prop<<<gEw, blk, 0, stream>>>(ei, ei + E, deg, bufA, bufB, E);
  lgcn_accum<<<gEl, blk, 0, stream>>>(bufB, alpha, 2, oute, bufA, n64);
  // Layer 3: bufB -> bufA ; out += a3*bufA
  lgcn_prop<<<gEw, blk, 0, stream>>>(ei, ei + E, deg, bufB, bufA, E);
  lgcn_accum<<<gEl, blk, 0, stream>>>(bufA, alpha, 3, oute, (float*)nullptr, n64);

  // Label dot products (WMMA f32 16x16x4, diagonal extraction)
  lgcn_dot_wmma<<<gD, blk, 0, stream>>>(eli, oute, (float*)d_out, L);
}